// LightGCNLayer_36249523978327
// MI455X (gfx1250) — compile-verified
//
#include <hip/hip_runtime.h>

typedef __attribute__((ext_vector_type(2))) float v2f;
typedef __attribute__((ext_vector_type(8))) float v8f;

#define DIM 128

#ifdef __has_builtin
#if __has_builtin(__builtin_amdgcn_wmma_f32_16x16x4_f32)
#define HAVE_WMMA_F32X4 1
#endif
#endif

// ---------------- CSR construction ----------------

__global__ void zero_counts(int* cnt_u, int* cnt_v, int* cursor, int U, int I, int N) {
    int i = blockIdx.x * blockDim.x + threadIdx.x;
    if (i < U) cnt_u[i] = 0;
    if (i < I) cnt_v[i] = 0;
    if (i < N) cursor[i] = 0;
}

__global__ void count_deg(const int* __restrict__ u_id, const int* __restrict__ v_id,
                          int* cnt_u, int* cnt_v, int E) {
    int e = blockIdx.x * blockDim.x + threadIdx.x;
    if (e < E) {
        atomicAdd(&cnt_u[u_id[e]], 1);
        atomicAdd(&cnt_v[v_id[e]], 1);
    }
}

__global__ void make_recip(const int* __restrict__ cnt_u, const int* __restrict__ cnt_v,
                           float* ru, float* rv, int U, int I) {
    int i = blockIdx.x * blockDim.x + threadIdx.x;
    if (i < U) ru[i] = 1.0f / (float)(cnt_u[i] + 1);   // u_deg = count + 1 (self loop)
    if (i < I) rv[i] = 1.0f / (float)(cnt_v[i] + 1);
}

// Exclusive scan of per-row entry counts (rounded up to a multiple of 4 so the
// SpMM loop needs no tail handling and quad loads stay 16B-aligned).
__global__ void scan_rowptr(const int* __restrict__ cnt_u, const int* __restrict__ cnt_v,
                            int* row_ptr, int U, int I, int N) {
    __shared__ int buf[1024];
    __shared__ int carry;
    if (threadIdx.x == 0) carry = 0;
    __syncthreads();
    for (int base = 0; base < N; base += 1024) {
        int r = base + (int)threadIdx.x;
        int c = 0;
        if (r < N) {
            c = 1;                      // diag entry
            if (r < U) c += cnt_u[r];   // edges with u_id == r
            if (r < I) c += cnt_v[r];   // edges with v_id == r (reference aliasing)
            c = (c + 3) & ~3;           // pad row to multiple of 4 entries
        }
        buf[threadIdx.x] = c;
        __syncthreads();
        for (int off = 1; off < 1024; off <<= 1) {
            int t = (threadIdx.x >= (unsigned)off) ? buf[threadIdx.x - off] : 0;
            __syncthreads();
            buf[threadIdx.x] += t;
            __syncthreads();
        }
        if (r < N) row_ptr[r] = carry + buf[threadIdx.x] - c;   // exclusive
        __syncthreads();
        if (threadIdx.x == 1023) carry += buf[1023];
        __syncthreads();
    }
    if (threadIdx.x == 0) row_ptr[N] = carry;
}

// Entries are {int col; float val} pairs (8B). Values precomputed at fill time
// so the SpMM loop does zero type decoding.
__global__ void fill_edges(const int* __restrict__ u_id, const int* __restrict__ v_id,
                           const int* __restrict__ row_ptr,
                           const float* __restrict__ ru, const float* __restrict__ rv,
                           int* cursor, int2* ent, int E) {
    int e = blockIdx.x * blockDim.x + threadIdx.x;
    if (e < E) {
        int u = u_id[e], v = v_id[e];
        int2 t0; t0.x = v; t0.y = __float_as_int(ru[u]);   // row u <- col v, 1/u_deg[u]
        int p0 = row_ptr[u] + atomicAdd(&cursor[u], 1);
        ent[p0] = t0;
        int2 t1; t1.x = u; t1.y = __float_as_int(rv[v]);   // row v <- col u, 1/v_deg[v]
        int p1 = row_ptr[v] + atomicAdd(&cursor[v], 1);
        ent[p1] = t1;
    }
}

__global__ void fill_diag(const int* __restrict__ row_ptr,
                          const float* __restrict__ ru, const float* __restrict__ rv,
                          int* cursor, int2* ent, int U, int N) {
    int r = blockIdx.x * blockDim.x + threadIdx.x;
    if (r < N) {
        float vd = (r < U) ? ru[r] : rv[r - U];
        int2 t; t.x = r; t.y = __float_as_int(vd);
        int p = row_ptr[r] + atomicAdd(&cursor[r], 1);
        ent[p] = t;
    }
}

// Padding slots: col = r (same cachelines as the row's diag entry -> ~free
// fetches), val = 0.0f (contributes nothing).
__global__ void pad_rows(const int* __restrict__ row_ptr, const int* __restrict__ cursor,
                         int2* ent, int N) {
    int r = blockIdx.x * blockDim.x + threadIdx.x;
    if (r < N) {
        int p = row_ptr[r] + cursor[r];
        int e = row_ptr[r + 1];
        int2 z; z.x = r; z.y = 0;
        for (; p < e; ++p) ent[p] = z;
    }
}

// ---------------- embeddings ----------------

__global__ void init_emb(const float* __restrict__ ue, const float* __restrict__ ie,
                         float* emb, float* outsum, int UD, int total) {
    int i = blockIdx.x * blockDim.x + threadIdx.x;
    if (i < total) {
        float x = (i < UD) ? ue[i] : ie[i - UD];
        emb[i] = x;      // layer-0 embedding
        outsum[i] = x;   // running sum for the final mean
    }
}

// ---------------- WMMA SpMM ----------------
// One wave = one CSR row, all 128 dims (8 accumulators, 8 WMMAs per quad).
// Per quad of neighbors (k0..k3): A[i][k] = val_k for ALL i (rank-degenerate A)
//   -> every D row of V_WMMA_F32_16X16X4_F32 = sum_k val_k * emb[c_k][16t+j].
// A layout (ISA 7.12.2, 32-bit 16x4): VGPR0 = K0(lanes0-15)/K2(lanes16-31), VGPR1 = K1/K3;
// B mirrors the same K convention. Row id is readfirstlane'd so loop control /
// row_ptr / entry loads are scalar (SALU + s_load); EXEC stays all-1s through
// the WMMA region. Entry quads are software-pipelined one iteration ahead.
// Final layer fuses the mean scale (accum = (accum + res) * scale).
__global__ void __launch_bounds__(256) spmm_layer(
        const float* __restrict__ ein, float* __restrict__ eout,
        float* __restrict__ accum, const int* __restrict__ row_ptr,
        const int4* __restrict__ entq /* 2 entries per int4 */, int N, float scale) {
    int wid = blockIdx.x * (blockDim.x >> 5) + ((int)threadIdx.x >> 5);
    int r = __builtin_amdgcn_readfirstlane(wid);   // wave-uniform row id (SGPR)
    if (r >= N) return;                            // scalar branch
    int lane = threadIdx.x & 31;
    int half = lane >> 4;
    int ln   = lane & 15;

    int start = __builtin_amdgcn_readfirstlane(row_ptr[r]);     // multiple of 4
    int end   = __builtin_amdgcn_readfirstlane(row_ptr[r + 1]);

    v8f acc[8] = {};

    // prologue: first quad (2x int4 = {c0,v0,c1,v1},{c2,v2,c3,v3}) -- s_load
    int4 ea = entq[start >> 1];
    int4 eb = entq[(start >> 1) + 1];

    for (int k = start; k < end; k += 4) {
        // prefetch next quad (32B overread slack allocated past the array)
        int nq = (k + 4) >> 1;
        int4 na = entq[nq];
        int4 nb = entq[nq + 1];

        // half 0 uses (k0,k1), half 1 uses (k2,k3) -- matches A/B K layout
        int   cx = half ? eb.x : ea.x;
        int   cy = half ? eb.z : ea.z;
        v2f A;
        A.x = __int_as_float(half ? eb.y : ea.y);
        A.y = __int_as_float(half ? eb.w : ea.w);

        const float* px = ein + cx * DIM + ln;   // 16 loads share 2 addr regs,
        const float* py = ein + cy * DIM + ln;   // tiles via immediate offsets
#pragma unroll
        for (int t = 0; t < 8; ++t) {
            v2f B;
            B.x = px[t * 16];
            B.y = py[t * 16];
#ifdef HAVE_WMMA_F32X4
            acc[t] = __builtin_amdgcn_wmma_f32_16x16x4_f32(
                /*neg_a=*/false, A, /*neg_b=*/false, B,
                /*c_mod=*/(short)0, acc[t], /*reuse_a=*/false, /*reuse_b=*/false);
#else
            acc[t][0] += A.x * B.x + A.y * B.y;  // per-half partial (host fallback)
#endif
        }
        ea = na;
        eb = nb;
    }

    // D row M=0 lives in acc[t][0]: lanes 0-15 hold cols 16t+ln.
    float res[8];
#pragma unroll
    for (int t = 0; t < 8; ++t) {
        float x = acc[t][0];
#ifndef HAVE_WMMA_F32X4
        x += __shfl_xor(x, 16);                  // combine half partials (fallback only)
#endif
        res[t] = x;
    }
    if (half == 0) {
        float* po = eout  + r * DIM + ln;
        float* pa = accum + r * DIM + ln;
#pragma unroll
        for (int t = 0; t < 8; ++t) {
            po[t * 16] = res[t];
            pa[t * 16] = (pa[t * 16] + res[t]) * scale;  // wave exclusively owns row
        }
    }
}

// ---------------- launch ----------------

extern "C" void kernel_launch(void* const* d_in, const int* in_sizes, int n_in,
                              void* d_out, int out_size, void* d_ws, size_t ws_size,
                              hipStream_t stream) {
    const float* ue  = (const float*)d_in[0];
    const float* ie  = (const float*)d_in[1];
    const int* u_id  = (const int*)d_in[2];
    const int* v_id  = (const int*)d_in[3];
    float* out = (float*)d_out;

    const int D = DIM;
    int U = in_sizes[0] / D;
    int I = in_sizes[1] / D;
    int E = in_sizes[2];
    int N = U + I;
    // worst case: 2E edge entries + N diag + up to 3 pad per row, + 64B slack for the 32B overread
    long long nnz_max = 2LL * E + 4LL * N + 8;

    char* ws = (char*)d_ws;
    size_t off = 0;
    auto carve = [&](size_t bytes) -> void* {
        void* p = ws + off;
        off += (bytes + 255) & ~(size_t)255;
        return p;
    };
    int*   cnt_u   = (int*)  carve((size_t)U * 4);
    int*   cnt_v   = (int*)  carve((size_t)I * 4);
    int*   cursor  = (int*)  carve((size_t)N * 4);
    int*   row_ptr = (int*)  carve((size_t)(N + 1) * 4);
    float* ru      = (float*)carve((size_t)U * 4);
    float* rv      = (float*)carve((size_t)I * 4);
    int2*  ent     = (int2*) carve((size_t)nnz_max * 8);
    float* emb_a   = (float*)carve((size_t)N * D * 4);
    float* emb_b   = (float*)carve((size_t)N * D * 4);
    (void)ws_size; (void)n_in; (void)out_size;

    const int thr = 256;
    zero_counts<<<(N + thr - 1) / thr, thr, 0, stream>>>(cnt_u, cnt_v, cursor, U, I, N);
    count_deg  <<<(E + thr - 1) / thr, thr, 0, stream>>>(u_id, v_id, cnt_u, cnt_v, E);
    make_recip <<<(U + thr - 1) / thr, thr, 0, stream>>>(cnt_u, cnt_v, ru, rv, U, I);
    scan_rowptr<<<1, 1024, 0, stream>>>(cnt_u, cnt_v, row_ptr, U, I, N);
    fill_edges <<<(E + thr - 1) / thr, thr, 0, stream>>>(u_id, v_id, row_ptr, ru, rv,
                                                         cursor, ent, E);
    fill_diag  <<<(N + thr - 1) / thr, thr, 0, stream>>>(row_ptr, ru, rv, cursor, ent, U, N);
    pad_rows   <<<(N + thr - 1) / thr, thr, 0, stream>>>(row_ptr, cursor, ent, N);

    int total = N * D;
    init_emb<<<(total + thr - 1) / thr, thr, 0, stream>>>(ue, ie, emb_a, out, U * D, total);

    int wpb = 256 / 32;                       // 8 rows (waves) per block
    int spmm_blocks = (N + wpb - 1) / wpb;
    const int4* entq = (const int4*)ent;
    spmm_layer<<<spmm_blocks, 256, 0, stream>>>(emb_a, emb_b, out, row_ptr, entq, N, 1.0f);
    spmm_layer<<<spmm_blocks, 256, 0, stream>>>(emb_b, emb_a, out, row_ptr, entq, N, 1.0f);
    spmm_layer<<<spmm_blocks, 256, 0, stream>>>(emb_a, emb_b, out, row_ptr, entq, N, 0.25f);
}